// HeteroDotProductPredictor_8332236554760
// MI455X (gfx1250) — compile-verified
//
#include <hip/hip_runtime.h>
#include <math.h>

// CDNA5 / gfx1250, wave32.
typedef float v2f __attribute__((ext_vector_type(2)));
typedef float v4f __attribute__((ext_vector_type(4)));
typedef float v8f __attribute__((ext_vector_type(8)));

#define FEAT_D 128

// ---------------------------------------------------------------------------
// Kernel 1: reciprocal L2 norm per row. One wave32 per 128-float row:
// each lane loads a float4 (b128), squares+accumulates, 5-step xor reduce,
// lane 0 writes 1/sqrt(sum).
// ---------------------------------------------------------------------------
__global__ __launch_bounds__(256) void rnorm_kernel(const float* __restrict__ h,
                                                    float* __restrict__ rn,
                                                    int nrows) {
  const int wave = (blockIdx.x * blockDim.x + threadIdx.x) >> 5;
  const int lane = threadIdx.x & 31;
  if (wave >= nrows) return;  // uniform per wave

  const v4f v = *(const v4f*)(h + (size_t)wave * FEAT_D + lane * 4);
  float s = v.x * v.x + v.y * v.y + v.z * v.z + v.w * v.w;
#pragma unroll
  for (int off = 16; off > 0; off >>= 1) s += __shfl_xor(s, off, 32);
  if (lane == 0) rn[wave] = 1.0f / sqrtf(s);
}

// ---------------------------------------------------------------------------
// Kernel 2: 16 edges per wave via V_WMMA_F32_16X16X4_F32.
//
// ISA §7.12.2 fp32 operand layouts (wave32):
//   A 16x4 : lane L -> row M = L&15 ; VGPR r -> K = (L>>4)*2 + r
//   B 4x16 : lane L -> col N = L&15 ; VGPR r -> K = (L>>4)*2 + r
// So for both operands, lane L needs TWO CONSECUTIVE features of its gathered
// row. We load a float4 per matrix per outer step (features kk*8 + h*4 .. +3,
// h = L>>4) and feed two WMMAs from it; the K->feature mapping is identical
// for A and B lanes, so the accumulated 16x16 product's diagonal is exactly
// the 16 edge dot products over all 128 features.
//
// D 16x16 : lane L (0..15)  VGPR r holds (M=r,   N=L)
//           lane L (16..31) VGPR r holds (M=r+8, N=L-16)
// Diagonal N==M lives in lanes {0..7} VGPR=lane and lanes {24..31} VGPR=lane-24.
// Those lanes' (lane&15) equals N, so they already hold the right src/dst.
// ---------------------------------------------------------------------------
__global__ __launch_bounds__(256) void edge_cos_wmma_kernel(
    const float* __restrict__ hg, const float* __restrict__ hd,
    const int* __restrict__ src, const int* __restrict__ dst,
    const float* __restrict__ rng, const float* __restrict__ rnd,
    float* __restrict__ out, int E) {
  const int wave = (blockIdx.x * blockDim.x + threadIdx.x) >> 5;
  const int lane = threadIdx.x & 31;
  const int e_base = wave * 16;
  if (e_base >= E) return;  // uniform per wave; EXEC stays all-1s for WMMA

  const int m = lane & 15;   // edge slot this lane serves
  const int h = lane >> 4;   // K half
  int e_m = e_base + m;
  const int e_c = (e_m < E) ? e_m : (E - 1);  // clamp, keep all lanes active

  const int si = src[e_c];
  const int di = dst[e_c];
  const float* gp = hg + (size_t)si * FEAT_D;
  const float* dp = hd + (size_t)di * FEAT_D;

  v8f c = {0.f, 0.f, 0.f, 0.f, 0.f, 0.f, 0.f, 0.f};

#pragma unroll 4
  for (int kk = 0; kk < 16; ++kk) {
    const int off = kk * 8 + h * 4;  // 16B aligned: full b128 gather per lane
    const v4f a4 = *(const v4f*)(gp + off);
    const v4f b4 = *(const v4f*)(dp + off);
    const v2f a0 = {a4.x, a4.y}, b0 = {b4.x, b4.y};
    const v2f a1 = {a4.z, a4.w}, b1 = {b4.z, b4.w};
    // (neg_a, A, neg_b, B, c_mod, C, reuse_a, reuse_b)
    c = __builtin_amdgcn_wmma_f32_16x16x4_f32(false, a0, false, b0,
                                              (short)0, c, false, false);
    c = __builtin_amdgcn_wmma_f32_16x16x4_f32(false, a1, false, b1,
                                              (short)0, c, false, false);
  }

  // Extract diagonal element for this lane (valid in lanes 0..7 and 24..31).
  const int sel = lane & 7;
  float d = c[0];
  if (sel == 1) d = c[1];
  if (sel == 2) d = c[2];
  if (sel == 3) d = c[3];
  if (sel == 4) d = c[4];
  if (sel == 5) d = c[5];
  if (sel == 6) d = c[6];
  if (sel == 7) d = c[7];

  const bool diag_lane = (lane < 8) || (lane >= 24);
  if (diag_lane && (e_m < E)) {
    out[e_m] = d * rng[si] * rnd[di];
  }
}

// ---------------------------------------------------------------------------
// Launch
// ---------------------------------------------------------------------------
extern "C" void kernel_launch(void* const* d_in, const int* in_sizes, int n_in,
                              void* d_out, int out_size, void* d_ws, size_t ws_size,
                              hipStream_t stream) {
  const float* hg = (const float*)d_in[0];
  const float* hd = (const float*)d_in[1];
  const int* src = (const int*)d_in[2];  // harness contract: integer -> const int*
  const int* dst = (const int*)d_in[3];
  float* out = (float*)d_out;

  const int Ng = in_sizes[0] / FEAT_D;  // 50000
  const int Nd = in_sizes[1] / FEAT_D;  // 20000
  const int E = in_sizes[2];            // 1500000

  float* rng = (float*)d_ws;            // Ng floats
  float* rnd = rng + Ng;                // Nd floats (total 280 KB << ws)

  // 8 waves per 256-thread block, one wave per row.
  rnorm_kernel<<<(Ng + 7) / 8, 256, 0, stream>>>(hg, rng, Ng);
  rnorm_kernel<<<(Nd + 7) / 8, 256, 0, stream>>>(hd, rnd, Nd);

  // One wave per 16 edges, 8 waves (128 edges) per block.
  const int waves = (E + 15) / 16;
  const int blocks = (waves + 7) / 8;
  edge_cos_wmma_kernel<<<blocks, 256, 0, stream>>>(hg, hd, src, dst, rng, rnd,
                                                   out, E);
}